// ParticleCollision_37855841747209
// MI455X (gfx1250) — compile-verified
//
#include <hip/hip_runtime.h>
#include <hip/hip_bf16.h>

// Problem constants (reference fixes these)
#define PC_B 2
#define PC_N 8192
#define PC_M 8192
#define PC_C 64
#define PC_K 128
#define PC_RADIUS 0.1f
#define PC_R2 0.01f
#define PC_MAXGD 96

typedef __attribute__((ext_vector_type(2))) float v2f;
typedef __attribute__((ext_vector_type(8))) float v8f;

static __device__ __forceinline__ unsigned ballot32(bool p) {
  return __builtin_amdgcn_ballot_w32(p);
}

// ---------------------------------------------------------------------------
// K1: per-batch min/max reduction over locs [B,N,3] -> stats[b] = {lo[3],_,up[3],_}
// ---------------------------------------------------------------------------
__global__ void __launch_bounds__(256)
pc_minmax_kernel(const float* __restrict__ locs, float* __restrict__ stats, int N) {
  __shared__ float smn[256 * 3];
  __shared__ float smx[256 * 3];
  const int b = blockIdx.x;
  const int t = threadIdx.x;
  float mn[3] = {1e30f, 1e30f, 1e30f};
  float mx[3] = {-1e30f, -1e30f, -1e30f};
  const float* base = locs + (size_t)b * N * 3;
  for (int i = t; i < N; i += 256) {
    #pragma unroll
    for (int d = 0; d < 3; ++d) {
      float v = base[(size_t)i * 3 + d];
      mn[d] = fminf(mn[d], v);
      mx[d] = fmaxf(mx[d], v);
    }
  }
  #pragma unroll
  for (int d = 0; d < 3; ++d) { smn[t * 3 + d] = mn[d]; smx[t * 3 + d] = mx[d]; }
  __syncthreads();
  for (int s = 128; s > 0; s >>= 1) {
    if (t < s) {
      #pragma unroll
      for (int d = 0; d < 3; ++d) {
        smn[t * 3 + d] = fminf(smn[t * 3 + d], smn[(t + s) * 3 + d]);
        smx[t * 3 + d] = fmaxf(smx[t * 3 + d], smx[(t + s) * 3 + d]);
      }
    }
    __syncthreads();
  }
  if (t == 0) {
    #pragma unroll
    for (int d = 0; d < 3; ++d) {
      stats[b * 8 + d]     = smn[d];
      stats[b * 8 + 4 + d] = smx[d];
    }
  }
}

// ---------------------------------------------------------------------------
// K2: per-batch stable argsort by flattened hash-grid cell id.
// 64-bit composite key (cid<<13 | i) bitonic-sorted entirely in LDS (64KB).
// ---------------------------------------------------------------------------
__global__ void __launch_bounds__(1024)
pc_sort_kernel(const float* __restrict__ locs, const float* __restrict__ stats,
               int* __restrict__ idxs, int N) {
  extern __shared__ unsigned long long keys[];  // N entries = 64KB
  const int b = blockIdx.x;
  const int t = threadIdx.x;

  float lo[3], up[3];
  int gd[3];
  #pragma unroll
  for (int d = 0; d < 3; ++d) {
    lo[d] = stats[b * 8 + d];
    up[d] = stats[b * 8 + 4 + d];
    int g = (int)floorf((up[d] - lo[d]) / PC_RADIUS) + 1;
    gd[d] = min(max(g, 1), PC_MAXGD);
  }
  const long long s2 = 1, s1 = gd[2], s0 = (long long)gd[1] * gd[2];

  const float* base = locs + (size_t)b * N * 3;
  for (int i = t; i < N; i += 1024) {
    long long cid = 0;
    const long long strides[3] = {s0, s1, s2};
    #pragma unroll
    for (int d = 0; d < 3; ++d) {
      float v = base[(size_t)i * 3 + d];
      int c = (int)floorf((v - lo[d]) / PC_RADIUS);
      c = min(max(c, 0), gd[d] - 1);
      cid += (long long)c * strides[d];
    }
    keys[i] = ((unsigned long long)cid << 13) | (unsigned)i;
  }

  for (unsigned k = 2; k <= (unsigned)N; k <<= 1) {
    for (unsigned j = k >> 1; j > 0; j >>= 1) {
      __syncthreads();
      #pragma unroll
      for (int e = 0; e < PC_N / 1024; ++e) {
        unsigned i = (unsigned)t + (unsigned)e * 1024u;
        unsigned ixj = i ^ j;
        if (ixj > i) {
          bool up_dir = ((i & k) == 0);
          unsigned long long a = keys[i];
          unsigned long long c = keys[ixj];
          if ((a > c) == up_dir) { keys[i] = c; keys[ixj] = a; }
        }
      }
    }
  }
  __syncthreads();
  for (int i = t; i < N; i += 1024)
    idxs[(size_t)b * N + i] = (int)(keys[i] & 8191ULL);
}

// ---------------------------------------------------------------------------
// K3: gather nlocs/ndata through idxs; precompute WMMA operands:
//   parr[b][j] = (-2px,-2py,-2pz, |p|^2)  from reordered locs
//   qarr[b][m] = ( qx,  qy,  qz,  |q|^2)
// ---------------------------------------------------------------------------
__global__ void __launch_bounds__(256)
pc_gather_kernel(const float* __restrict__ locs, const float* __restrict__ data,
                 const float* __restrict__ qlocs, const int* __restrict__ idxs,
                 float* __restrict__ nlocs, float* __restrict__ ndata,
                 float4* __restrict__ parr, float4* __restrict__ qarr,
                 int N, int M, int C) {
  const int tid = blockIdx.x * 256 + threadIdx.x;
  if (tid < PC_B * N) {
    const int b = tid / N;
    const int src = idxs[tid];
    const float* lp = locs + ((size_t)b * N + src) * 3;
    float x = lp[0], y = lp[1], z = lp[2];
    float* np = nlocs + (size_t)tid * 3;
    np[0] = x; np[1] = y; np[2] = z;
    parr[tid] = make_float4(-2.0f * x, -2.0f * y, -2.0f * z, x * x + y * y + z * z);
    const float4* ds = (const float4*)(data + ((size_t)b * N + src) * C);
    float4* dd = (float4*)(ndata + (size_t)tid * C);
    #pragma unroll
    for (int k = 0; k < PC_C / 4; ++k) dd[k] = ds[k];
  }
  if (tid < PC_B * M) {
    const float* qp = qlocs + (size_t)tid * 3;
    float x = qp[0], y = qp[1], z = qp[2];
    qarr[tid] = make_float4(x, y, z, x * x + y * y + z * z);
  }
}

// ---------------------------------------------------------------------------
// K4: collision tiles via V_WMMA_F32_16X16X4_F32.
//   A row m = (qx,qy,qz,1), B col n = (-2p, |p|^2), C[m][n] = |q_m|^2 - r^2
//   => D[m][n] = d^2(m,n) - r^2 ; hit iff D <= 0.
// Ballot per C-VGPR -> two 16-bit row masks; an 8-way uniform select chain
// routes ballot word (lane&7) to each owner lane; ctz append keeps ascending
// order. Particle operands staged into 128KB LDS via
// GLOBAL_LOAD_ASYNC_TO_LDS_B128 (ASYNCcnt-tracked); tile stream pipelined.
// ---------------------------------------------------------------------------
__global__ void __launch_bounds__(256)
pc_collide_kernel(const float4* __restrict__ parr, const float4* __restrict__ qarr,
                  int* __restrict__ nbr, int N, int M) {
  extern __shared__ float4 pl[];  // N float4 = 128KB
  const int tid = threadIdx.x;
  const int groups = M / 128;               // query groups per batch
  const int b = blockIdx.x / groups;
  const int g = blockIdx.x % groups;

  // Async global->LDS staging (CDNA5 async path, tracked by ASYNCcnt).
  {
    const float4* gsrc = parr + (size_t)b * N;
    for (int i = tid; i < N; i += 256) {
      unsigned ldsb = (unsigned)(uintptr_t)(pl + i);              // addr[31:0] = LDS offset
      unsigned long long ga = (unsigned long long)(uintptr_t)(gsrc + i);
      asm volatile("global_load_async_to_lds_b128 %0, %1, off"
                   :: "v"(ldsb), "v"(ga) : "memory");
    }
    asm volatile("s_wait_asynccnt 0x0" ::: "memory");
  }
  __syncthreads();

  const int lane = tid & 31;
  const int w = tid >> 5;                   // wave id: 16 queries per wave
  const int jn = lane & 15;
  const int qrow = g * 128 + w * 16 + jn;

  float4 qv = qarr[(size_t)b * M + qrow];
  v2f a;
  a.x = (lane < 16) ? qv.x : qv.z;          // A vgpr0: K=0 | K=2
  a.y = (lane < 16) ? qv.y : 1.0f;          // A vgpr1: K=1 | K=3 (pairs |p|^2)

  // Seed C with |q_m|^2 - r^2 distributed in the 16x16 f32 C layout:
  // lane l, vgpr v -> row (v + 8*(l>=16)).
  float negthr = qv.w - PC_R2;
  v8f cinit;
  #pragma unroll
  for (int v = 0; v < 8; ++v)
    cinit[v] = __shfl(negthr, v + ((lane & 16) >> 1), 32);

  int* nbout = nbr + ((size_t)b * M + (size_t)(g * 128 + w * 16 + jn)) * PC_K;
  int cnt = 0;

  float4 pv = pl[jn];                       // pipelined tile operand
  for (int jt = 0; jt < N; jt += 16) {
    const int jnext = (jt + 16 < N) ? (jt + 16) : 0;
    float4 pnx = pl[jnext + jn];            // prefetch next tile (ds_load overlaps WMMA)

    v2f bb;
    bb.x = (lane < 16) ? pv.x : pv.z;       // B vgpr0: K=0 | K=2
    bb.y = (lane < 16) ? pv.y : pv.w;       // B vgpr1: K=1 | K=3

    v8f c = __builtin_amdgcn_wmma_f32_16x16x4_f32(
        /*neg_a=*/false, a, /*neg_b=*/false, bb,
        /*c_mod=*/(short)0, cinit, /*reuse_a=*/false, /*reuse_b=*/false);

    unsigned bal[8];
    #pragma unroll
    for (int v = 0; v < 8; ++v)
      bal[v] = ballot32(c[v] <= 0.0f);      // low16 = row v, high16 = row v+8

    unsigned any = bal[0] | bal[1] | bal[2] | bal[3] |
                   bal[4] | bal[5] | bal[6] | bal[7];
    if (any) {
      // Route uniform ballot word (lane&7) to each lane: 8-way select chain
      // over uniform (SGPR) values, generated once for the whole wave.
      const unsigned sel = (unsigned)(lane & 7);
      unsigned rm32 = bal[0];
      #pragma unroll
      for (int v = 1; v < 8; ++v)
        rm32 = (sel == (unsigned)v) ? bal[v] : rm32;
      if (lane < 16) {
        unsigned rm = (lane < 8) ? (rm32 & 0xffffu) : (rm32 >> 16);
        while (rm) {
          int bit = __builtin_ctz(rm);
          rm &= rm - 1;
          if (cnt < PC_K) nbout[cnt] = jt + bit;
          ++cnt;
        }
      }
    }
    pv = pnx;
  }
  if (lane < 16) {
    for (int k = cnt; k < PC_K; ++k) nbout[k] = -1;
  }
}

// ---------------------------------------------------------------------------
// Launcher
// ---------------------------------------------------------------------------
extern "C" void kernel_launch(void* const* d_in, const int* in_sizes, int n_in,
                              void* d_out, int out_size, void* d_ws, size_t ws_size,
                              hipStream_t stream) {
  const float* locs  = (const float*)d_in[0];   // [B,N,3]
  const float* data  = (const float*)d_in[1];   // [B,N,64]
  const float* qlocs = (const float*)d_in[2];   // [B,M,3]

  const int B = PC_B, N = PC_N, M = PC_M, C = PC_C;

  // d_out layout (4-byte units): nlocs | ndata | idxs | neighbors
  float* out_f = (float*)d_out;
  float* nlocs = out_f;                               // B*N*3
  float* ndata = out_f + (size_t)B * N * 3;           // B*N*C
  int*   idxs  = (int*)(out_f + (size_t)B * N * 3 + (size_t)B * N * C);   // B*N
  int*   nbr   = idxs + (size_t)B * N;                                     // B*M*128

  // workspace layout
  float*  stats = (float*)d_ws;                             // B*8 floats
  float4* parr  = (float4*)((char*)d_ws + 256);             // B*N float4
  float4* qarr  = parr + (size_t)B * N;                     // B*M float4

  (void)hipFuncSetAttribute((const void*)pc_sort_kernel,
                            hipFuncAttributeMaxDynamicSharedMemorySize, PC_N * 8);
  (void)hipFuncSetAttribute((const void*)pc_collide_kernel,
                            hipFuncAttributeMaxDynamicSharedMemorySize, PC_N * 16);

  pc_minmax_kernel<<<B, 256, 0, stream>>>(locs, stats, N);
  pc_sort_kernel<<<B, 1024, PC_N * 8, stream>>>(locs, stats, idxs, N);

  int gather_threads = B * ((N > M) ? N : M);
  pc_gather_kernel<<<(gather_threads + 255) / 256, 256, 0, stream>>>(
      locs, data, qlocs, idxs, nlocs, ndata, parr, qarr, N, M, C);

  pc_collide_kernel<<<B * (M / 128), 256, PC_N * 16, stream>>>(parr, qarr, nbr, N, M);
}